// Syn_28930899706245
// MI455X (gfx1250) — compile-verified
//
#include <hip/hip_runtime.h>

// ---------------------------------------------------------------------------
// Spiking-network synaptic update for MI455X (gfx1250).
// Phase 1: elementwise neuron update (Ieff -> out, r_new -> workspace).
// Phase 2: COO scatter-add. Edge/weight stream (~120 MB, read-once) is DMA'd
//          into LDS via GLOBAL_LOAD_ASYNC_TO_LDS_B128 with th:TH_LOAD_NT so
//          it does not churn the 192 MB L2; the r_new gathers (2 MB) and
//          global_atomic_add_f32 scatter (2 MB) stay L2-resident.
// Roofline: ~130 MB HBM @ 23.3 TB/s -> ~6 us floor; FLOPs negligible.
// ---------------------------------------------------------------------------

constexpr int   TILE            = 128;  // edges per wave per tile (4 per lane)
constexpr int   WAVES_PER_BLOCK = 8;    // 256 threads, wave32
constexpr int   EDGE_BLOCKS     = 2048;

constexpr float DT      = 0.1f;
constexpr float TAO_D   = 2.0f;
constexpr float TAO_R   = 10.0f;
constexpr float LAMDA_D = DT / TAO_D;   // 0.05
constexpr float LAMDA_R = DT / TAO_R;   // 0.01

// -------------------------- phase 1: neurons -------------------------------
__global__ void __launch_bounds__(256)
neuron_kernel(const float* __restrict__ Iback,
              const float* __restrict__ spike,
              const float* __restrict__ s,
              const float* __restrict__ r,
              const float* __restrict__ noise,
              const float* __restrict__ dt_over_tau,
              const float* __restrict__ sqrt_coeff,
              const float* __restrict__ sig,
              const float* __restrict__ mu,
              float* __restrict__ out,
              float* __restrict__ r_new_ws,
              int n)
{
    const float a  = dt_over_tau[0];
    const float sc = sqrt_coeff[0];
    const float sg = sig[0];
    const float m  = mu[0];

    int i = blockIdx.x * blockDim.x + threadIdx.x;
    if (i >= n) return;

    float ib   = Iback[i];
    float ibn  = ib + a * (noise[i] - ib);
    float ieff = ibn / sc * sg + m;

    float sv   = s[i];
    float snew = sv + LAMDA_R * (-sv + spike[i] * (1.0f / TAO_D));
    float rv   = r[i];
    float rnew = rv - LAMDA_D * rv + DT * snew;

    out[i]       = ieff;   // edge kernel accumulates on top of Ieff
    r_new_ws[i]  = rnew;
}

// -------------------------- async helpers ----------------------------------
// Low 32 bits of the flat address of a __shared__ object are the LDS offset
// (flat LDS aperture: addr[63:32]=SHARED_BASE, addr[31:0]=LDS offset).
__device__ __forceinline__ unsigned ldsOff(const void* p) {
    return (unsigned)(size_t)p;
}

// GVS form: global_load_async_to_lds_b128 vdst(lds addr), vaddr(u32 off), saddr
// th:TH_LOAD_NT -> non-temporal near+far: read-once stream, keep out of L2.
__device__ __forceinline__ void asyncLoadB128NT(unsigned lds, unsigned goff,
                                                const void* base) {
    asm volatile("global_load_async_to_lds_b128 %0, %1, %2 th:TH_LOAD_NT"
                 :
                 : "v"(lds), "v"(goff),
                   "s"((unsigned long long)(size_t)base)
                 : "memory");
}

__device__ __forceinline__ void waitAsyncLe3() {
    asm volatile("s_wait_asynccnt 3" ::: "memory");
}
__device__ __forceinline__ void waitAsyncLe0() {
    asm volatile("s_wait_asynccnt 0" ::: "memory");
}

// -------------------------- phase 2: edges ---------------------------------
__global__ void __launch_bounds__(256)
edge_kernel(const int*   __restrict__ post,
            const int*   __restrict__ pre,
            const float* __restrict__ w,
            const float* __restrict__ rnew,
            float* __restrict__ out,
            int nEdge, int nFullTiles, int tilesPerWave)
{
    // Double-buffered per-wave slices: 2 * 8 waves * 128 * 12B = 24 KB LDS.
    __shared__ __align__(16) int   sPost[2][WAVES_PER_BLOCK][TILE];
    __shared__ __align__(16) int   sPre [2][WAVES_PER_BLOCK][TILE];
    __shared__ __align__(16) float sW   [2][WAVES_PER_BLOCK][TILE];

    const int lane = threadIdx.x & 31;
    const int wv   = threadIdx.x >> 5;
    const long long waveId = (long long)blockIdx.x * WAVES_PER_BLOCK + wv;

    long long tBeg = waveId * (long long)tilesPerWave;
    long long tEnd = tBeg + tilesPerWave;
    if (tEnd > nFullTiles) tEnd = nFullTiles;

    if (tBeg < tEnd) {
        const int li = lane * 4;  // 16B-aligned slot inside the wave slice
        {   // prime buffer 0
            unsigned eo = (unsigned)((tBeg * TILE + li) * 4);
            asyncLoadB128NT(ldsOff(&sPost[0][wv][li]), eo, post);
            asyncLoadB128NT(ldsOff(&sPre [0][wv][li]), eo, pre);
            asyncLoadB128NT(ldsOff(&sW   [0][wv][li]), eo, w);
        }
        int buf = 0;
        for (long long t = tBeg; t < tEnd; ++t) {
            if (t + 1 < tEnd) {   // prefetch next tile into the other buffer
                unsigned eo = (unsigned)(((t + 1) * TILE + li) * 4);
                int nb = buf ^ 1;
                asyncLoadB128NT(ldsOff(&sPost[nb][wv][li]), eo, post);
                asyncLoadB128NT(ldsOff(&sPre [nb][wv][li]), eo, pre);
                asyncLoadB128NT(ldsOff(&sW   [nb][wv][li]), eo, w);
                waitAsyncLe3();   // async loads retire in order:
                                  // cnt<=3 => current tile's 3 are done
            } else {
                waitAsyncLe0();
            }

            int4   p  = *reinterpret_cast<const int4*>(&sPost[buf][wv][li]);
            int4   q  = *reinterpret_cast<const int4*>(&sPre [buf][wv][li]);
            float4 ww = *reinterpret_cast<const float4*>(&sW  [buf][wv][li]);

            // L2-resident random gathers (r_new is 2 MB << 192 MB L2)
            float r0 = rnew[q.x];
            float r1 = rnew[q.y];
            float r2 = rnew[q.z];
            float r3 = rnew[q.w];

            // no-return global_atomic_add_f32 (STOREcnt tracked)
            atomicAdd(&out[p.x], ww.x * r0);
            atomicAdd(&out[p.y], ww.y * r1);
            atomicAdd(&out[p.z], ww.z * r2);
            atomicAdd(&out[p.w], ww.w * r3);

            buf ^= 1;
        }
    }

    // Scalar tail for edges not covered by full tiles (empty when 128 | nEdge).
    long long remStart = (long long)nFullTiles * TILE;
    long long gid    = (long long)blockIdx.x * blockDim.x + threadIdx.x;
    long long stride = (long long)gridDim.x * blockDim.x;
    for (long long e = remStart + gid; e < nEdge; e += stride) {
        atomicAdd(&out[post[e]], w[e] * rnew[pre[e]]);
    }
}

// ---------------------------------------------------------------------------
extern "C" void kernel_launch(void* const* d_in, const int* in_sizes, int n_in,
                              void* d_out, int out_size, void* d_ws, size_t ws_size,
                              hipStream_t stream)
{
    const float* weight      = (const float*)d_in[0];
    const int*   edges       = (const int*)  d_in[1];  // row0=post, row1=pre
    const float* Iback       = (const float*)d_in[2];
    const float* spike       = (const float*)d_in[3];
    const float* s           = (const float*)d_in[4];
    const float* r           = (const float*)d_in[5];
    const float* noise       = (const float*)d_in[6];
    const float* dt_over_tau = (const float*)d_in[7];
    const float* sqrt_coeff  = (const float*)d_in[8];
    const float* sig         = (const float*)d_in[9];
    const float* mu          = (const float*)d_in[10];

    float* out = (float*)d_out;
    float* rws = (float*)d_ws;   // r_new scratch: nNeuron floats

    const int nNeuron = in_sizes[2];
    const int nEdge   = in_sizes[1] / 2;
    const int* post = edges;
    const int* pre  = edges + nEdge;

    dim3 blk(256);
    dim3 grd1((nNeuron + 255) / 256);
    neuron_kernel<<<grd1, blk, 0, stream>>>(Iback, spike, s, r, noise,
                                            dt_over_tau, sqrt_coeff, sig, mu,
                                            out, rws, nNeuron);

    const int nFullTiles = nEdge / TILE;                     // 78125 for 10M
    const long long totalWaves = (long long)EDGE_BLOCKS * WAVES_PER_BLOCK;
    int tilesPerWave = (int)((nFullTiles + totalWaves - 1) / totalWaves);
    if (tilesPerWave < 1) tilesPerWave = 1;

    edge_kernel<<<dim3(EDGE_BLOCKS), blk, 0, stream>>>(post, pre, weight, rws,
                                                       out, nEdge, nFullTiles,
                                                       tilesPerWave);
}